// HybridKernelTransformer_65481071403107
// MI455X (gfx1250) — compile-verified
//
#include <hip/hip_runtime.h>
#include <math.h>

// Problem constants (from reference)
#define EDIM  1024
#define NWA   8
#define FFDIM 4096
#define NWF   4
#define NBLK  6
#define CDIM  1000
#define BATCH 4096

typedef float v2f __attribute__((ext_vector_type(2)));
typedef float v8f __attribute__((ext_vector_type(8)));

// ---------------------------------------------------------------------------
// WMMA matvec helper: accumulate D(16x16) += A(16x4) * B(4x16) over K range.
// A = 16 consecutive weight rows (row-major, stride = ldk elements).
// B = vector replicated across all 16 N columns => every D column equals W*vec.
// Lane layout (ISA 7.12.2): lanes 0-15 hold M=lane, K={0,1}; lanes 16-31 hold
// M=lane-16, K={2,3}. So per lane: koff = (lane<16)?0:2, m = lane&15.
// ---------------------------------------------------------------------------
__device__ __forceinline__ v8f wmma_matvec(const float* __restrict__ arow,
                                           const float* vec,
                                           int kbeg, int kend, int koff)
{
    v8f acc = {0.f, 0.f, 0.f, 0.f, 0.f, 0.f, 0.f, 0.f};
    __builtin_prefetch(arow + kbeg, 0, 3);
    for (int k = kbeg; k < kend; k += 4) {
        v2f a; a.x = arow[k + koff]; a.y = arow[k + koff + 1];
        v2f b; b.x = vec[k + koff];  b.y = vec[k + koff + 1];
        acc = __builtin_amdgcn_wmma_f32_16x16x4_f32(
            /*neg_a=*/false, a, /*neg_b=*/false, b,
            /*c_mod=*/(short)0, acc, /*reuse_a=*/false, /*reuse_b=*/false);
    }
    return acc;
}

// D rows 0-7 live in lanes 0-15 (all copies identical), rows 8-15 in lanes 16-31.
__device__ __forceinline__ void stash_rows(float red[4][16], int wv, int lane, v8f acc)
{
    if (lane == 0) {
        #pragma unroll
        for (int r = 0; r < 8; ++r) red[wv][r] = acc[r];
    } else if (lane == 16) {
        #pragma unroll
        for (int r = 0; r < 8; ++r) red[wv][8 + r] = acc[r];
    }
}

// ---------------------------------------------------------------------------
// h init: kern == 1 exactly, pe0 = [sin(0),cos(0),...] = [0,1,0,1,...]
// ---------------------------------------------------------------------------
__global__ __launch_bounds__(EDIM) void init_h_kernel(float* __restrict__ h)
{
    int e = threadIdx.x;
    h[e] = (e & 1) ? 2.0f : 1.0f;
}

// ---------------------------------------------------------------------------
// Attention (S=1 => softmax==1 => o=v):  u = h + Wo * (Wv * cos(h[0:8]+phi))
// grid 64 blocks x 128 threads; block b owns output rows [16b, 16b+16).
// 4 waves split K=1024 into 256-wide quarters, LDS-reduce partials.
// ---------------------------------------------------------------------------
__global__ __launch_bounds__(128) void attn_kernel(const float* __restrict__ h,
                                                   const float* __restrict__ Wv,
                                                   const float* __restrict__ Wo,
                                                   const float* __restrict__ phi,
                                                   float* __restrict__ u)
{
    __shared__ float projs[NWA];
    __shared__ float vsh[EDIM];
    __shared__ float red[4][16];

    const int tid = threadIdx.x;
    if (tid < NWA) projs[tid] = cosf(h[tid] + phi[tid]);
    __syncthreads();

    for (int e = tid; e < EDIM; e += 128) {
        const float* w = Wv + (size_t)e * NWA;
        float s = 0.f;
        #pragma unroll
        for (int j = 0; j < NWA; ++j) s += projs[j] * w[j];
        vsh[e] = s;
    }
    __syncthreads();

    const int lane = tid & 31;
    const int wv   = tid >> 5;
    const int m    = lane & 15;
    const int koff = (lane < 16) ? 0 : 2;
    const int row  = blockIdx.x * 16 + m;

    v8f acc = wmma_matvec(Wo + (size_t)row * EDIM, vsh,
                          wv * 256, wv * 256 + 256, koff);
    stash_rows(red, wv, lane, acc);
    __syncthreads();

    if (tid < 16) {
        float s = red[0][tid] + red[1][tid] + red[2][tid] + red[3][tid];
        int e = blockIdx.x * 16 + tid;
        u[e] = h[e] + s;
    }
}

// ---------------------------------------------------------------------------
// FFN:  u = h + W2 * relu(W1 * (cos(h[0:4]) * cos(phi_f)))
// grid 64 blocks x 128 threads; r[4096] staged in LDS (16 KB).
// ---------------------------------------------------------------------------
__global__ __launch_bounds__(128) void ffn_kernel(const float* __restrict__ h,
                                                  const float* __restrict__ W1,
                                                  const float* __restrict__ W2,
                                                  const float* __restrict__ phi,
                                                  float* __restrict__ u)
{
    __shared__ float msh[NWF];
    __shared__ float rsh[FFDIM];
    __shared__ float red[4][16];

    const int tid = threadIdx.x;
    if (tid < NWF) msh[tid] = cosf(h[tid]) * cosf(phi[tid]);
    __syncthreads();

    for (int ff = tid; ff < FFDIM; ff += 128) {
        const float* w = W1 + (size_t)ff * NWF;
        float s = msh[0] * w[0] + msh[1] * w[1] + msh[2] * w[2] + msh[3] * w[3];
        rsh[ff] = fmaxf(s, 0.f);
    }
    __syncthreads();

    const int lane = tid & 31;
    const int wv   = tid >> 5;
    const int m    = lane & 15;
    const int koff = (lane < 16) ? 0 : 2;
    const int row  = blockIdx.x * 16 + m;

    v8f acc = wmma_matvec(W2 + (size_t)row * FFDIM, rsh,
                          wv * 1024, wv * 1024 + 1024, koff);
    stash_rows(red, wv, lane, acc);
    __syncthreads();

    if (tid < 16) {
        float s = red[0][tid] + red[1][tid] + red[2][tid] + red[3][tid];
        int e = blockIdx.x * 16 + tid;
        u[e] = h[e] + s;
    }
}

// ---------------------------------------------------------------------------
// LayerNorm over E=1024: one block of 1024 threads (32 waves, wave32 shuffles).
// ---------------------------------------------------------------------------
__global__ __launch_bounds__(EDIM) void ln_kernel(const float* __restrict__ u,
                                                  const float* __restrict__ g,
                                                  const float* __restrict__ b,
                                                  float* __restrict__ hout)
{
    __shared__ float ssum[32];
    __shared__ float ssq[32];
    __shared__ float smean, srstd;

    const int tid = threadIdx.x;
    float x = u[tid];
    float s = x, q = x * x;
    #pragma unroll
    for (int off = 16; off > 0; off >>= 1) {
        s += __shfl_xor(s, off, 32);
        q += __shfl_xor(q, off, 32);
    }
    if ((tid & 31) == 0) { ssum[tid >> 5] = s; ssq[tid >> 5] = q; }
    __syncthreads();
    if (tid < 32) {
        s = ssum[tid]; q = ssq[tid];
        #pragma unroll
        for (int off = 16; off > 0; off >>= 1) {
            s += __shfl_xor(s, off, 32);
            q += __shfl_xor(q, off, 32);
        }
        if (tid == 0) {
            float mean = s * (1.0f / EDIM);
            float var  = q * (1.0f / EDIM) - mean * mean;
            smean = mean;
            srstd = rsqrtf(var + 1e-5f);
        }
    }
    __syncthreads();
    hout[tid] = (x - smean) * srstd * g[tid] + b[tid];
}

// ---------------------------------------------------------------------------
// Classifier: logits[c] = dot(Wc[c], h) + bc[c], c < 1000.
// 63 blocks of 16 rows (1008 >= 1000; loads clamped, stores guarded).
// ---------------------------------------------------------------------------
__global__ __launch_bounds__(128) void cls_kernel(const float* __restrict__ h,
                                                  const float* __restrict__ Wc,
                                                  const float* __restrict__ bc,
                                                  float* __restrict__ logits)
{
    __shared__ float hsh[EDIM];
    __shared__ float red[4][16];

    const int tid = threadIdx.x;
    for (int e = tid; e < EDIM; e += 128) hsh[e] = h[e];
    __syncthreads();

    const int lane = tid & 31;
    const int wv   = tid >> 5;
    const int m    = lane & 15;
    const int koff = (lane < 16) ? 0 : 2;
    int row = blockIdx.x * 16 + m;
    if (row >= CDIM) row = CDIM - 1;   // clamp A loads in bounds

    v8f acc = wmma_matvec(Wc + (size_t)row * EDIM, hsh,
                          wv * 256, wv * 256 + 256, koff);
    stash_rows(red, wv, lane, acc);
    __syncthreads();

    if (tid < 16) {
        int c = blockIdx.x * 16 + tid;
        if (c < CDIM) {
            float s = red[0][tid] + red[1][tid] + red[2][tid] + red[3][tid];
            logits[c] = s + bc[c];
        }
    }
}

// ---------------------------------------------------------------------------
// Broadcast: every batch row is identical. Write-bound (16.4 MB).
// grid (4, 4096) x 256: block.y = batch row, coalesced 1000-float row writes.
// ---------------------------------------------------------------------------
__global__ __launch_bounds__(256) void bcast_kernel(const float* __restrict__ logits,
                                                    float* __restrict__ out)
{
    int c = blockIdx.x * 256 + threadIdx.x;
    if (c < CDIM) {
        out[(size_t)blockIdx.y * CDIM + c] = logits[c];
    }
}

// ---------------------------------------------------------------------------
// Launch: h -> 6x [attn, LN1, ffn, LN2] -> classifier -> broadcast.
// Inputs (setup_inputs order): 0 x, 1 Wq, 2 Wk, 3 Wv, 4 Wo, 5 phi_q, 6 W1,
// 7 W2, 8 phi_f, 9 ln1_g, 10 ln1_b, 11 ln2_g, 12 ln2_b, 13 Wc, 14 bc.
// x/Wq/Wk unused (dead in the reference math).
// ---------------------------------------------------------------------------
extern "C" void kernel_launch(void* const* d_in, const int* in_sizes, int n_in,
                              void* d_out, int out_size, void* d_ws, size_t ws_size,
                              hipStream_t stream)
{
    (void)in_sizes; (void)n_in; (void)out_size; (void)ws_size;

    const float* Wv    = (const float*)d_in[3];
    const float* Wo    = (const float*)d_in[4];
    const float* phi_q = (const float*)d_in[5];
    const float* W1    = (const float*)d_in[6];
    const float* W2    = (const float*)d_in[7];
    const float* phi_f = (const float*)d_in[8];
    const float* ln1_g = (const float*)d_in[9];
    const float* ln1_b = (const float*)d_in[10];
    const float* ln2_g = (const float*)d_in[11];
    const float* ln2_b = (const float*)d_in[12];
    const float* Wc    = (const float*)d_in[13];
    const float* bc    = (const float*)d_in[14];
    float* out = (float*)d_out;

    float* h      = (float*)d_ws;     // EDIM
    float* u      = h + EDIM;         // EDIM
    float* logits = u + EDIM;         // CDIM (padded)

    init_h_kernel<<<1, EDIM, 0, stream>>>(h);

    for (int i = 0; i < NBLK; ++i) {
        attn_kernel<<<EDIM / 16, 128, 0, stream>>>(
            h, Wv + (size_t)i * EDIM * NWA, Wo + (size_t)i * EDIM * EDIM,
            phi_q + i * NWA, u);
        ln_kernel<<<1, EDIM, 0, stream>>>(u, ln1_g + i * EDIM, ln1_b + i * EDIM, h);
        ffn_kernel<<<EDIM / 16, 128, 0, stream>>>(
            h, W1 + (size_t)i * FFDIM * NWF, W2 + (size_t)i * EDIM * FFDIM,
            phi_f + i * NWF, u);
        ln_kernel<<<1, EDIM, 0, stream>>>(u, ln2_g + i * EDIM, ln2_b + i * EDIM, h);
    }

    cls_kernel<<<(CDIM + 15) / 16, 128, 0, stream>>>(h, Wc, bc, logits);
    bcast_kernel<<<dim3(4, BATCH), 256, 0, stream>>>(logits, out);
}